// Model_33268816675314
// MI455X (gfx1250) — compile-verified
//
#include <hip/hip_runtime.h>

#define B_ 32
#define H_ 8
#define S_ 384
#define S1_ 385
#define K_ 16
#define D_ 256
#define HD_ 32
#define GW_EPS_ 0.05f
#define GW_SIGMA_ 1.0f
#define OUTER_ 5
#define SINK_ 20
#define EPS_ 1e-8f

typedef __attribute__((ext_vector_type(2))) float v2f;
typedef __attribute__((ext_vector_type(8))) float v8f;

__device__ __forceinline__ v8f wmma_f32(v2f a, v2f b, v8f c) {
  // V_WMMA_F32_16X16X4_F32 : D = A(16x4) * B(4x16) + C(16x16)
  return __builtin_amdgcn_wmma_f32_16x16x4_f32(
      false, a, false, b, (short)0, c, false, false);
}

// ---------------------------------------------------------------------------
// Kernel 1: per-(b,h) entropic GW -> alpha[b,h]
// One block (256 threads, 8 waves) per (b,h).
// ---------------------------------------------------------------------------
__global__ __launch_bounds__(256) void gw_kernel(
    const float* __restrict__ shared_attn,  // [B,H,385,385]
    const float* __restrict__ DG,           // [B,H,16,16]
    const float* __restrict__ bvec,         // [B,H,16]
    float* __restrict__ alpha_ws)           // [B*H]
{
  __shared__ float sT[S_ * K_];     // transport plan T
  __shared__ float sK[S_ * K_];     // M buffer then Kmat
  __shared__ float sA[S_];          // marginal a
  __shared__ float sU[S_];          // sinkhorn u
  __shared__ float sCc1[S_];        // cC1
  __shared__ float sC2[K_ * K_];    // DG slice
  __shared__ float sCc2[K_];
  __shared__ float sB[K_];
  __shared__ float sV[K_];
  __shared__ float sColT[K_];
  __shared__ float sRed[256];

  const int tid = threadIdx.x;
  const int bh = blockIdx.x;
  const float* P = shared_attn + (size_t)bh * S1_ * S1_;  // P_var(i,j) = P[(1+i)*385 + 1+j]

  if (tid < K_) sB[tid] = bvec[bh * K_ + tid];
  { int k = tid >> 4, l = tid & 15;
    sC2[k * K_ + l] = DG[(size_t)bh * K_ * K_ + k * K_ + l]; }

  // ---- Phase A: rowsum/colsum/max -> mx, a ----
  float m_loc = -1e30f, degsum_loc = 0.f;
  for (int j = tid; j < S_; j += 256) {
    float rs = 0.f, cs = 0.f, mj = -1e30f;
    const float* rowj = P + (size_t)(1 + j) * S1_ + 1;
    for (int i = 0; i < S_; ++i) {
      float pij = P[(size_t)(1 + i) * S1_ + 1 + j];  // coalesced across lanes
      float pji = rowj[i];
      cs += pij; rs += pji;
      mj = fmaxf(mj, 0.5f * (pij + pji));
    }
    float deg = 0.5f * (rs + cs);
    sA[j] = deg;
    degsum_loc += deg;
    m_loc = fmaxf(m_loc, mj);
  }
  sRed[tid] = m_loc; __syncthreads();
  for (int s = 128; s > 0; s >>= 1) { if (tid < s) sRed[tid] = fmaxf(sRed[tid], sRed[tid + s]); __syncthreads(); }
  const float mxc = fmaxf(sRed[0], EPS_);
  __syncthreads();
  sRed[tid] = degsum_loc; __syncthreads();
  for (int s = 128; s > 0; s >>= 1) { if (tid < s) sRed[tid] += sRed[tid + s]; __syncthreads(); }
  const float dsum = fmaxf(sRed[0] / mxc, EPS_);
  __syncthreads();
  const float ainv = 1.0f / (mxc * dsum);
  for (int j = tid; j < S_; j += 256) sA[j] *= ainv;  // a[j]
  __syncthreads();

  // ---- Phase B: cC1, cC2, T init ----
  const float halfinv = 0.5f / mxc;
  for (int i = tid; i < S_; i += 256) {
    float acc = 0.f;
    const float* rowi = P + (size_t)(1 + i) * S1_ + 1;
    for (int j = 0; j < S_; ++j) {
      float pij = rowi[j];
      float pji = P[(size_t)(1 + j) * S1_ + 1 + i];
      float dp = 1.0f - (pij + pji) * halfinv;
      acc += dp * dp * sA[j];
    }
    sCc1[i] = acc;
  }
  if (tid < K_) {
    float acc = 0.f;
    for (int k = 0; k < K_; ++k) { float c = sC2[k * K_ + tid]; acc += sB[k] * c * c; }
    sCc2[tid] = acc;
  }
  __syncthreads();
  for (int idx = tid; idx < S_ * K_; idx += 256) {
    int i = idx >> 4, l = idx & 15;
    sT[idx] = sA[i] * sB[l];
  }
  __syncthreads();

  // ---- Phase C: GW outer loop ----
  const float inv_eps = 1.0f / GW_EPS_;
  const int wave = tid >> 5;
  const int lane = tid & 31;
  const int row = lane & 15;
  const int kb = (lane >> 4) << 1;  // 0 or 2 (f32 WMMA A/B layout)
  const int col = lane & 15;
  const int rbase = (lane >> 4) << 3;

  for (int outer = 0; outer < OUTER_; ++outer) {
    // colT[k] = sum_i T[i,k]
    {
      int k = tid & 15, chunk = tid >> 4;
      float p = 0.f;
      for (int i = chunk * 24; i < chunk * 24 + 24; ++i) p += sT[i * K_ + k];
      sRed[tid] = p;
      __syncthreads();
      if (tid < K_) {
        float s = 0.f;
        for (int g = 0; g < 16; ++g) s += sRed[g * K_ + tid];
        sColT[tid] = s;
      }
      __syncthreads();
    }
    // M = C1@T = colT - halfinv*(P@T + P^T@T)  (WMMA, 3 i-tiles per wave)
    for (int t = 0; t < 3; ++t) {
      const int i0 = (wave + t * 8) * 16;
      v8f acc = {0.f, 0.f, 0.f, 0.f, 0.f, 0.f, 0.f, 0.f};
      const float* rowA = P + (size_t)(1 + i0 + row) * S1_ + 1;
      for (int j0 = 0; j0 < S_; j0 += 4) {
        v2f a1, a2, bf;
        a1.x = rowA[j0 + kb];
        a1.y = rowA[j0 + kb + 1];
        a2.x = P[(size_t)(1 + j0 + kb) * S1_ + 1 + i0 + row];      // transposed tile
        a2.y = P[(size_t)(1 + j0 + kb + 1) * S1_ + 1 + i0 + row];
        bf.x = sT[(j0 + kb) * K_ + col];
        bf.y = sT[(j0 + kb + 1) * K_ + col];
        acc = wmma_f32(a1, bf, acc);
        acc = wmma_f32(a2, bf, acc);
      }
#pragma unroll
      for (int e = 0; e < 8; ++e) {
        int r = e + rbase;
        sK[(i0 + r) * K_ + col] = sColT[col] - halfinv * acc[e];
      }
    }
    // tens = const - 2*M@C2^T ; Kmat = exp(-tens/eps)  (per-wave rows only)
    for (int t = 0; t < 3; ++t) {
      const int i0 = (wave + t * 8) * 16;
      v8f acc = {0.f, 0.f, 0.f, 0.f, 0.f, 0.f, 0.f, 0.f};
      for (int k0 = 0; k0 < K_; k0 += 4) {
        v2f a, bf;
        a.x = sK[(i0 + row) * K_ + k0 + kb];
        a.y = sK[(i0 + row) * K_ + k0 + kb + 1];
        bf.x = sC2[col * K_ + k0 + kb];       // C2^T[k,col] = C2[col,k]
        bf.y = sC2[col * K_ + k0 + kb + 1];
        acc = wmma_f32(a, bf, acc);
      }
#pragma unroll
      for (int e = 0; e < 8; ++e) {
        int r = e + rbase;
        float tv = sCc1[i0 + r] + sCc2[col] - 2.0f * acc[e];
        sK[(i0 + r) * K_ + col] = __expf(-tv * inv_eps);
      }
    }
    __syncthreads();
    // Sinkhorn
    if (tid < K_) sV[tid] = 1.0f;
    __syncthreads();
    for (int it = 0; it < SINK_; ++it) {
      for (int i = tid; i < S_; i += 256) {
        float r = 0.f;
#pragma unroll
        for (int k = 0; k < K_; ++k) r += sK[i * K_ + k] * sV[k];
        sU[i] = sA[i] / fmaxf(r, 1e-30f);
      }
      __syncthreads();
      {
        int k = tid & 15, chunk = tid >> 4;
        float p = 0.f;
        for (int i = chunk * 24; i < chunk * 24 + 24; ++i) p += sK[i * K_ + k] * sU[i];
        sRed[tid] = p;
      }
      __syncthreads();
      if (tid < K_) {
        float c = 0.f;
        for (int g = 0; g < 16; ++g) c += sRed[g * K_ + tid];
        sV[tid] = sB[tid] / fmaxf(c, 1e-30f);
      }
      __syncthreads();
    }
    // T = u * K * v
    for (int idx = tid; idx < S_ * K_; idx += 256) {
      int i = idx >> 4, k = idx & 15;
      sT[idx] = sU[i] * sK[idx] * sV[k];
    }
    __syncthreads();
  }

  // gw = sum(tens * T), tens = -eps*ln(Kmat)
  float g_loc = 0.f;
  for (int idx = tid; idx < S_ * K_; idx += 256)
    g_loc += (-GW_EPS_ * __logf(fmaxf(sK[idx], 1e-35f))) * sT[idx];
  sRed[tid] = g_loc; __syncthreads();
  for (int s = 128; s > 0; s >>= 1) { if (tid < s) sRed[tid] += sRed[tid + s]; __syncthreads(); }
  if (tid == 0) alpha_ws[bh] = __expf(-sRed[0] / GW_SIGMA_);
}

// ---------------------------------------------------------------------------
// Kernel 2: sharpen prior_Q -> attn (one wave per output row)
// ---------------------------------------------------------------------------
__global__ __launch_bounds__(256) void attn_kernel(
    const float* __restrict__ prior_Q,   // [B,H,384,384]
    const float* __restrict__ alpha_ws,  // [B*H]
    float* __restrict__ attn_out)        // [B,H,385,385]
{
  const int gw = blockIdx.x * 8 + (threadIdx.x >> 5);
  const int lane = threadIdx.x & 31;
  const int i = gw % S1_;
  const int bh = gw / S1_;
  float* out = attn_out + (size_t)bh * S1_ * S1_ + (size_t)i * S1_;
  if (i == 0) {
    for (int c = lane; c < S1_; c += 32) out[c] = (c == 0) ? 0.f : (1.0f / 384.0f);
    return;
  }
  const float e = 1.0f + alpha_ws[bh];
  const float* q = prior_Q + (size_t)bh * S_ * S_ + (size_t)(i - 1) * S_;
  float p[12];
  float sum = 0.f, diag = 0.f;
#pragma unroll
  for (int k = 0; k < 12; ++k) {
    int j = lane + k * 32;
    float val = __expf(e * __logf(fmaxf(q[j], EPS_)));
    p[k] = val;
    sum += val;
    if (j == i - 1) diag = val;
  }
#pragma unroll
  for (int m = 16; m >= 1; m >>= 1) {
    sum += __shfl_xor(sum, m, 32);
    diag += __shfl_xor(diag, m, 32);
  }
  const float r1 = 1.0f / fmaxf(sum, EPS_);
  const float ssum = (sum - diag) * r1;
  const float scale = r1 / fmaxf(ssum, EPS_);
  if (lane == 0) out[0] = 0.f;
#pragma unroll
  for (int k = 0; k < 12; ++k) {
    int j = lane + k * 32;
    out[1 + j] = (j == i - 1) ? 0.f : p[k] * scale;
  }
}

// ---------------------------------------------------------------------------
// Kernel 3: v = nv @ v_w^T + v_b   (WMMA f32, one 16x16 tile per wave)
// rows = B*S1 = 12320 (divisible by 16), cols = 256
// ---------------------------------------------------------------------------
__global__ __launch_bounds__(256) void vproj_kernel(
    const float* __restrict__ nv, const float* __restrict__ vw,
    const float* __restrict__ vb, float* __restrict__ vout)
{
  const int wid = blockIdx.x * 8 + (threadIdx.x >> 5);
  const int lane = threadIdx.x & 31;
  const int mt = wid >> 4, nt = wid & 15;
  const int i0 = mt * 16, n0 = nt * 16;
  const int row = lane & 15;
  const int kb = (lane >> 4) << 1;
  const int col = lane & 15;
  const int rbase = (lane >> 4) << 3;
  v8f acc = {0.f, 0.f, 0.f, 0.f, 0.f, 0.f, 0.f, 0.f};
  const float* arow = nv + (size_t)(i0 + row) * D_;
  const float* brow = vw + (size_t)(n0 + col) * D_;  // B[k,n] = v_w[n,k]
  for (int k0 = 0; k0 < D_; k0 += 4) {
    v2f a, b;
    a.x = arow[k0 + kb]; a.y = arow[k0 + kb + 1];
    b.x = brow[k0 + kb]; b.y = brow[k0 + kb + 1];
    acc = wmma_f32(a, b, acc);
  }
  const float bias = vb[n0 + col];
#pragma unroll
  for (int e = 0; e < 8; ++e) {
    int r = e + rbase;
    vout[(size_t)(i0 + r) * D_ + n0 + col] = acc[e] + bias;
  }
}

// ---------------------------------------------------------------------------
// Kernel 4: context = attn @ v (per head), write transposed basis output
// per (b,h): [385,385] x [385,32]; M padded to 400, K=385 edge-masked
// ---------------------------------------------------------------------------
__global__ __launch_bounds__(256) void ctx_kernel(
    const float* __restrict__ attn, const float* __restrict__ v,
    float* __restrict__ basis)
{
  const int wid = blockIdx.x * 8 + (threadIdx.x >> 5);
  const int lane = threadIdx.x & 31;
  const int nt = wid & 1;
  const int mt = (wid >> 1) % 25;
  const int bh = wid / 50;
  const int bb = bh >> 3, h = bh & 7;
  const int i0 = mt * 16, n0 = nt * 16;
  const int row = lane & 15;
  const int kb = (lane >> 4) << 1;
  const int col = lane & 15;
  const int rbase = (lane >> 4) << 3;
  const float* A = attn + (size_t)bh * S1_ * S1_;
  const int ar = min(i0 + row, S1_ - 1);           // clamp M edge (store-guarded)
  const float* arow = A + (size_t)ar * S1_;
  v8f acc = {0.f, 0.f, 0.f, 0.f, 0.f, 0.f, 0.f, 0.f};
  for (int k0 = 0; k0 < S1_; k0 += 4) {
    const int kk0 = k0 + kb, kk1 = kk0 + 1;
    const int kc0 = min(kk0, S1_ - 1), kc1 = min(kk1, S1_ - 1);
    v2f a, b;
    a.x = arow[kc0];
    a.y = arow[kc1];
    float b0 = v[(size_t)(bb * S1_ + kc0) * D_ + h * HD_ + n0 + col];
    float b1 = v[(size_t)(bb * S1_ + kc1) * D_ + h * HD_ + n0 + col];
    b.x = (kk0 < S1_) ? b0 : 0.0f;                 // zero invalid K (masks A too)
    b.y = (kk1 < S1_) ? b1 : 0.0f;
    acc = wmma_f32(a, b, acc);
  }
#pragma unroll
  for (int e = 0; e < 8; ++e) {
    int r = i0 + e + rbase;
    if (r < S1_)
      basis[(((size_t)bb * S1_ + r) * H_ + h) * HD_ + n0 + col] = acc[e];
  }
}

// ---------------------------------------------------------------------------
extern "C" void kernel_launch(void* const* d_in, const int* in_sizes, int n_in,
                              void* d_out, int out_size, void* d_ws, size_t ws_size,
                              hipStream_t stream) {
  (void)in_sizes; (void)n_in; (void)out_size; (void)ws_size;
  // inputs: 0 desc(unused), 1 nv, 2 shared_attn, 3 prior_Q, 4 DG, 5 b, 6 v_w, 7 v_b
  const float* nv          = (const float*)d_in[1];
  const float* shared_attn = (const float*)d_in[2];
  const float* prior_Q     = (const float*)d_in[3];
  const float* DG          = (const float*)d_in[4];
  const float* bvec        = (const float*)d_in[5];
  const float* v_w         = (const float*)d_in[6];
  const float* v_b         = (const float*)d_in[7];

  float* basis_out = (float*)d_out;                              // [32,385,8,32]
  float* attn_out  = (float*)d_out + (size_t)B_ * S1_ * D_;      // [32,8,385,385]

  float* alpha_ws = (float*)d_ws;            // 256 floats
  float* v_ws     = alpha_ws + 256;          // [32,385,256] floats (12.6 MB)

  gw_kernel<<<B_ * H_, 256, 0, stream>>>(shared_attn, DG, bvec, alpha_ws);
  attn_kernel<<<(B_ * H_ * S1_) / 8, 256, 0, stream>>>(prior_Q, alpha_ws, attn_out);
  vproj_kernel<<<(770 * 16) / 8, 256, 0, stream>>>(nv, v_w, v_b, v_ws);
  ctx_kernel<<<(B_ * H_ * 25 * 2) / 8, 256, 0, stream>>>(attn_out, v_ws, basis_out);
}